// AgentImputer_64828236366127
// MI455X (gfx1250) — compile-verified
//
#include <hip/hip_runtime.h>
#include <math.h>

typedef __attribute__((ext_vector_type(2))) float v2f;
typedef __attribute__((ext_vector_type(8))) float v8f;
typedef __attribute__((ext_vector_type(4))) int   v4i;

#define B_    64
#define W_    128
#define N_    40
#define FIN_  66
#define HID_  100
#define NEWD_ 71
#define KPAD_ 72
#define G4_   400      // 4*HID
#define E_    240
#define R_    (B_*N_)        // 2560 sequences
#define ROWS_ (R_*W_)        // 327680 GEMM rows

#if defined(__has_builtin)
#if __has_builtin(__builtin_amdgcn_global_load_async_to_lds_b128)
#define HAVE_ASYNC_LDS 1
#endif
#if __has_builtin(__builtin_amdgcn_s_wait_asynccnt)
#define HAVE_WAIT_ASYNC 1
#endif
#endif

#ifdef HAVE_ASYNC_LDS
typedef __attribute__((address_space(1))) v4i gv4i;   // global int4
typedef __attribute__((address_space(3))) v4i lv4i;   // LDS int4
#endif

__device__ __forceinline__ void wait_async_zero() {
#ifdef HAVE_WAIT_ASYNC
  __builtin_amdgcn_s_wait_asynccnt(0);
#elif defined(__AMDGCN__)
  asm volatile("s_wait_asynccnt 0x0" ::: "memory");
#endif
}

// ---------------------------------------------------------------------------
// Kernel 1: build embedded features Xe (padded K=72) and transposed ts.
// Xe[row*72 + k], row = (b*N + n)*W + w   (matches xa reshape in reference)
// ---------------------------------------------------------------------------
__global__ void k_embed(const float* __restrict__ X, const float* __restrict__ ts,
                        const float* __restrict__ emb_pos, const float* __restrict__ emb_team,
                        float* __restrict__ Xe, float* __restrict__ tsa) {
  long long e = (long long)blockIdx.x * blockDim.x + threadIdx.x;
  long long total = (long long)ROWS_ * KPAD_;
  if (e >= total) return;
  int k = (int)(e % KPAD_);
  long long row = e / KPAD_;
  int w  = (int)(row % W_);
  int rn = (int)(row / W_);
  int n = rn % N_;
  int b = rn / N_;
  long long base = ((long long)(b * W_ + w) * N_ + n) * FIN_;
  float val;
  if (k < 64) {
    val = X[base + k];
  } else if (k < 68) {
    int ip = (int)X[base + 64];
    val = (ip > 0) ? emb_pos[(ip - 1) * 4 + (k - 64)] : 0.f;
  } else if (k < 71) {
    int it = (int)X[base + 65];
    val = (it > 0) ? emb_team[(it - 1) * 3 + (k - 68)] : 0.f;
  } else {
    val = 0.f;                                  // zero pad (safe for WMMA K loop)
    tsa[row] = ts[(long long)(b * W_ + w) * N_ + n];
  }
  Xe[row * KPAD_ + k] = val;
}

// ---------------------------------------------------------------------------
// Kernel 2: u = Xe @ Uall_w^T + Uall_b via V_WMMA_F32_16X16X4_F32.
// One 16x16 output tile per wave, K-loop of 18 WMMAs (K padded to 72).
// ---------------------------------------------------------------------------
__global__ void __launch_bounds__(256)
k_ugemm(const float* __restrict__ Xe, const float* __restrict__ Uall_w,
        const float* __restrict__ Uall_b, float* __restrict__ u) {
  const int NT = G4_ / 16;                       // 25 column tiles
  int wid  = blockIdx.x * (blockDim.x >> 5) + (threadIdx.x >> 5);
  int lane = threadIdx.x & 31;
  int half = lane >> 4;
  int l15  = lane & 15;
  int tm = wid / NT;
  int tn = wid % NT;
  long long m0 = (long long)tm * 16;
  int n0 = tn * 16;

  const float* arow = Xe + (m0 + l15) * KPAD_;   // A row for this lane (M = lane&15)
  int ncol = n0 + l15;
  const float* brow = Uall_w + (long long)ncol * NEWD_;  // B col = Uall_w row

  v8f acc = {};
  for (int kk = 0; kk < KPAD_; kk += 4) {
    int ka = kk + 2 * half;                      // per-lane K pair (even -> 8B aligned)
    __builtin_prefetch(arow + ka + 16, 0, 1);    // global_prefetch_b8
    v2f a = *(const v2f*)(arow + ka);
    v2f bb;
    bb.x = (ka     < NEWD_) ? brow[ka]     : 0.f;
    bb.y = (ka + 1 < NEWD_) ? brow[ka + 1] : 0.f;
    acc = __builtin_amdgcn_wmma_f32_16x16x4_f32(false, a, false, bb,
                                                (short)0, acc, false, false);
  }
  float bias = Uall_b[ncol];
  float* od = u + (m0 + 8 * half) * G4_ + ncol;  // D: VGPR r -> row r + 8*half
#pragma unroll
  for (int r = 0; r < 8; ++r)
    od[(long long)r * G4_] = acc[r] + bias;
}

// ---------------------------------------------------------------------------
// Kernel 3: persistent-weight time-decay LSTM scan + output projection.
// 16 sequences per workgroup, 4 waves; Wall^T/Wd^T/lin^T live in LDS.
// Per step: async-DMA u_s into LDS (overlapped with WMMA), 7 c_s1 tiles +
// 25 gate tiles of WMMA f32, then elementwise gate update.
// Total LDS = 320,768 B  (fits the 320KB WGP budget, 1 WG per WGP).
// ---------------------------------------------------------------------------
__global__ void __launch_bounds__(128)
k_scan(const float* __restrict__ u, const float* __restrict__ tsa,
       const float* __restrict__ Wall_w, const float* __restrict__ Wall_b,
       const float* __restrict__ Wd_w,   const float* __restrict__ Wd_b,
       const float* __restrict__ lin_w,  const float* __restrict__ lin_b,
       float* __restrict__ nodes) {
  extern __shared__ float smem[];
  float* sWall = smem;                 // [100][400]  Wall^T
  float* sWd   = sWall + 100 * 400;    // [100][112]  Wd^T  (N padded, zero cols)
  float* sLin  = sWd   + 100 * 112;    // [100][112]  lin^T
  float* sh    = sLin  + 100 * 112;    // [16][100]   h
  float* sc    = sh    + 16 * 100;     // [16][100]   c
  float* sg    = sc    + 16 * 100;     // [16][400]   gate pre-activations (no u)
  float* scs   = sg    + 16 * 400;     // [16][112]   c@Wd^T + b pre-tanh
  float* sU    = scs   + 16 * 112;     // [16][400]   async-staged u_s for step t

  int tid  = threadIdx.x;
  int wave = tid >> 5, lane = tid & 31, half = lane >> 4, l15 = lane & 15;
  int r0 = blockIdx.x * 16;            // first sequence of this block

  for (int i = tid; i < 100 * 400; i += blockDim.x) {
    int k = i / 400, n = i % 400;
    sWall[i] = Wall_w[n * 100 + k];
  }
  for (int i = tid; i < 100 * 112; i += blockDim.x) {
    int k = i / 112, n = i % 112;
    sWd[i]  = (n < 100) ? Wd_w[n * 100 + k]  : 0.f;
    sLin[i] = (n < 100) ? lin_w[n * 100 + k] : 0.f;
  }
  for (int i = tid; i < 16 * 100; i += blockDim.x) { sh[i] = 0.f; sc[i] = 0.f; }
  __syncthreads();

  for (int t = 0; t < W_; ++t) {
    // ---- stage u_s for this step into LDS; DMA overlaps the WMMA phase below.
    // sU was last read in the previous step's elementwise phase, which is
    // separated from here by a barrier, so overwrite is safe.
    for (int i = tid; i < 16 * (G4_ / 4); i += blockDim.x) {   // 1600 float4s
      int m = i / (G4_ / 4);
      int f = i % (G4_ / 4);
      const float* gp = u + ((long long)(r0 + m) * W_ + t) * G4_ + f * 4;
      float* lp = sU + m * G4_ + f * 4;
#ifdef HAVE_ASYNC_LDS
      __builtin_amdgcn_global_load_async_to_lds_b128(
          (gv4i*)gp, (lv4i*)lp, 0, 0);
#else
      *(float4*)lp = *(const float4*)gp;
#endif
    }

    // ---- matmul phase: tiles 0..6 = c_s1 (K=100,N=112), 7..31 = gates (N=400)
    for (int tile = wave; tile < 32; tile += 4) {
      bool isCs = tile < 7;
      int n0 = isCs ? tile * 16 : (tile - 7) * 16;
      const float* sA = isCs ? sc : sh;
      const float* sB = isCs ? sWd : sWall;
      int ldb = isCs ? 112 : 400;
      v8f acc = {};
      for (int kk = 0; kk < HID_; kk += 4) {
        int ka = kk + 2 * half;
        v2f a = *(const v2f*)(sA + l15 * HID_ + ka);
        v2f bb;
        bb.x = sB[ka * ldb + n0 + l15];
        bb.y = sB[(ka + 1) * ldb + n0 + l15];
        acc = __builtin_amdgcn_wmma_f32_16x16x4_f32(false, a, false, bb,
                                                    (short)0, acc, false, false);
      }
      int nn = n0 + l15;
      if (isCs) {
        float bias = (nn < HID_) ? Wd_b[nn] : 0.f;
#pragma unroll
        for (int r = 0; r < 8; ++r)
          scs[(r + 8 * half) * 112 + nn] = acc[r] + bias;
      } else {
        float bias = Wall_b[nn];
#pragma unroll
        for (int r = 0; r < 8; ++r)
          sg[(r + 8 * half) * 400 + nn] = acc[r] + bias;
      }
    }
    wait_async_zero();                 // own async loads landed in LDS
    __syncthreads();                   // ...and visible to all waves

    // ---- elementwise gate update (adds u_s from staged sU)
    for (int i = tid; i < 16 * HID_; i += blockDim.x) {
      int m = i / HID_, j = i % HID_;
      float tsv  = tsa[(long long)(r0 + m) * W_ + t];
      float cold = sc[i];
      float cs1  = tanhf(scs[m * 112 + j]);
      float cadj = cold - cs1 + cs1 * tsv;
      const float* ur = sU + m * 400;
      float f  = 1.f / (1.f + __expf(-(sg[m * 400 + j]       + ur[j])));
      float ig = 1.f / (1.f + __expf(-(sg[m * 400 + 100 + j] + ur[100 + j])));
      float o  = 1.f / (1.f + __expf(-(sg[m * 400 + 200 + j] + ur[200 + j])));
      float ct = 1.f / (1.f + __expf(-(sg[m * 400 + 300 + j] + ur[300 + j])));
      float cn = f * cadj + ig * ct;
      sc[i] = cn;
      sh[i] = o * tanhf(cn);
    }
    __syncthreads();
  }

  // ---- outs = relu(h_last @ lin^T + lin_b) -> nodes
  for (int tile = wave; tile < 7; tile += 4) {
    int n0 = tile * 16;
    v8f acc = {};
    for (int kk = 0; kk < HID_; kk += 4) {
      int ka = kk + 2 * half;
      v2f a = *(const v2f*)(sh + l15 * HID_ + ka);
      v2f bb;
      bb.x = sLin[ka * 112 + n0 + l15];
      bb.y = sLin[(ka + 1) * 112 + n0 + l15];
      acc = __builtin_amdgcn_wmma_f32_16x16x4_f32(false, a, false, bb,
                                                  (short)0, acc, false, false);
    }
    int nn = n0 + l15;
    if (nn < HID_) {
      float bias = lin_b[nn];
#pragma unroll
      for (int r = 0; r < 8; ++r) {
        float v = acc[r] + bias;
        nodes[(long long)(r0 + r + 8 * half) * HID_ + nn] = v > 0.f ? v : 0.f;
      }
    }
  }
}

// ---------------------------------------------------------------------------
// Kernel 4: two GraphSAGE layers + output head, one block per batch element.
// Tiny (N=40, E=240) -> plain VALU with everything in LDS.
// ---------------------------------------------------------------------------
__global__ void __launch_bounds__(128)
k_sage(const float* __restrict__ nodes, const int* __restrict__ ei,
       const float* __restrict__ s1l, const float* __restrict__ s1lb,
       const float* __restrict__ s1r,
       const float* __restrict__ s2l, const float* __restrict__ s2lb,
       const float* __restrict__ s2r,
       const float* __restrict__ ow,  const float* __restrict__ ob,
       float* __restrict__ out) {
  __shared__ float x[N_ * HID_];
  __shared__ float agg[N_ * HID_];
  __shared__ float g1[N_ * 64];
  __shared__ float agg1[N_ * 64];
  __shared__ float g2[N_ * 32];
  __shared__ float deg[N_];
  int b = blockIdx.x, tid = threadIdx.x;
  const int* src = ei;
  const int* dst = ei + E_;

  for (int i = tid; i < N_ * HID_; i += blockDim.x)
    x[i] = nodes[(long long)(b * N_ + i / HID_) * HID_ + i % HID_];
  if (tid < N_) {
    float d = 0.f;
    for (int e = 0; e < E_; ++e) d += (dst[e] == tid) ? 1.f : 0.f;
    deg[tid] = d > 1.f ? d : 1.f;               // clip(deg, 1.0)
  }
  __syncthreads();

  for (int i = tid; i < N_ * HID_; i += blockDim.x) {
    int n = i / HID_, f = i % HID_;
    float s = 0.f;
    for (int e = 0; e < E_; ++e)
      if (dst[e] == n) s += x[src[e] * HID_ + f];
    agg[i] = s / deg[n];
  }
  __syncthreads();

  for (int i = tid; i < N_ * 64; i += blockDim.x) {
    int n = i / 64, j = i % 64;
    float s = s1lb[j];
    for (int k = 0; k < HID_; ++k)
      s += agg[n * HID_ + k] * s1l[j * HID_ + k] + x[n * HID_ + k] * s1r[j * HID_ + k];
    g1[i] = s > 0.f ? s : 0.f;
  }
  __syncthreads();

  for (int i = tid; i < N_ * 64; i += blockDim.x) {
    int n = i / 64, j = i % 64;
    float s = 0.f;
    for (int e = 0; e < E_; ++e)
      if (dst[e] == n) s += g1[src[e] * 64 + j];
    agg1[i] = s / deg[n];
  }
  __syncthreads();

  for (int i = tid; i < N_ * 32; i += blockDim.x) {
    int n = i / 32, j = i % 32;
    float s = s2lb[j];
    for (int k = 0; k < 64; ++k)
      s += agg1[n * 64 + k] * s2l[j * 64 + k] + g1[n * 64 + k] * s2r[j * 64 + k];
    g2[i] = s > 0.f ? s : 0.f;
  }
  __syncthreads();

  for (int i = tid; i < N_ * 2; i += blockDim.x) {
    int n = i / 2, j = i % 2;
    float s = ob[j];
    for (int k = 0; k < 32; ++k) s += g2[n * 32 + k] * ow[j * 32 + k];
    out[(long long)(b * N_ + n) * 2 + j] = s > 0.f ? s : 0.f;
  }
}

// ---------------------------------------------------------------------------
extern "C" void kernel_launch(void* const* d_in, const int* in_sizes, int n_in,
                              void* d_out, int out_size, void* d_ws, size_t ws_size,
                              hipStream_t stream) {
  const float* X        = (const float*)d_in[0];
  const float* ts       = (const float*)d_in[1];
  const int*   ei       = (const int*)  d_in[2];
  const float* emb_pos  = (const float*)d_in[3];
  const float* emb_team = (const float*)d_in[4];
  const float* Wall_w   = (const float*)d_in[5];
  const float* Wall_b   = (const float*)d_in[6];
  const float* Uall_w   = (const float*)d_in[7];
  const float* Uall_b   = (const float*)d_in[8];
  const float* Wd_w     = (const float*)d_in[9];
  const float* Wd_b     = (const float*)d_in[10];
  const float* lin_w    = (const float*)d_in[11];
  const float* lin_b    = (const float*)d_in[12];
  const float* s1l      = (const float*)d_in[13];
  const float* s1lb     = (const float*)d_in[14];
  const float* s1r      = (const float*)d_in[15];
  const float* s2l      = (const float*)d_in[16];
  const float* s2lb     = (const float*)d_in[17];
  const float* s2r      = (const float*)d_in[18];
  const float* ow       = (const float*)d_in[19];
  const float* ob       = (const float*)d_in[20];
  float* out = (float*)d_out;
  float* ws  = (float*)d_ws;

  float* Xe    = ws;
  float* tsa   = Xe  + (size_t)ROWS_ * KPAD_;
  float* u     = tsa + (size_t)ROWS_;
  float* nodes = u   + (size_t)ROWS_ * G4_;

  {
    long long total = (long long)ROWS_ * KPAD_;
    int thr = 256;
    unsigned blocks = (unsigned)((total + thr - 1) / thr);
    k_embed<<<dim3(blocks), thr, 0, stream>>>(X, ts, emb_pos, emb_team, Xe, tsa);
  }
  {
    int tiles  = (ROWS_ / 16) * (G4_ / 16);      // 512000 output tiles
    int blocks = tiles / 8;                      // 8 waves (tiles) per 256-thread block
    k_ugemm<<<blocks, 256, 0, stream>>>(Xe, Uall_w, Uall_b, u);
  }
  {
    size_t lds = (size_t)(100 * 400 + 100 * 112 + 100 * 112 +
                          16 * 100 + 16 * 100 + 16 * 400 + 16 * 112 +
                          16 * 400) * sizeof(float);   // 320,768 B
    k_scan<<<R_ / 16, 128, lds, stream>>>(u, tsa, Wall_w, Wall_b, Wd_w, Wd_b,
                                          lin_w, lin_b, nodes);
  }
  k_sage<<<B_, 128, 0, stream>>>(nodes, ei, s1l, s1lb, s1r,
                                 s2l, s2lb, s2r, ow, ob, out);
}